// DiffeqSolver_augmented_26809185861945
// MI455X (gfx1250) — compile-verified
//
#include <hip/hip_runtime.h>
#include <math.h>

// ---- problem constants (from reference) ----
#define TRAJ   3
#define BATCH  2048
#define LAT    123
#define DAUG   128      // latents + AUG
#define HID    256
#define TSTEPS 50
#define MROWS  (TRAJ * BATCH)   // 6144

typedef float v2f __attribute__((ext_vector_type(2)));
typedef float v8f __attribute__((ext_vector_type(8)));

// LDS float offsets
#define OFF_W1   0                         // [128][256]  B1[k][n] = W1[n][k]
#define OFF_W2   (OFF_W1 + 128 * 256)      // [256][128]  B2[k][d] = W2[d][k]
#define OFF_B1   (OFF_W2 + 256 * 128)      // [256]
#define OFF_B2   (OFF_B1 + 256)            // [128]
#define OFF_Y    (OFF_B2 + 128)            // [16][129]
#define OFF_YIN  (OFF_Y   + 16 * 129)
#define OFF_K1   (OFF_YIN + 16 * 129)
#define OFF_K2   (OFF_K1  + 16 * 129)
#define OFF_K3   (OFF_K2  + 16 * 129)
#define OFF_H    (OFF_K3  + 16 * 129)      // [16][257]
#define OFF_TS   (OFF_H   + 16 * 257)      // [64]
#define LDS_FLOATS (OFF_TS + 64)

__global__ void __launch_bounds__(256)
DiffeqSolver_augmented_26809185861945_kernel(
    const float* __restrict__ fp,   // (3,2048,123)
    const float* __restrict__ ts,   // (50,)
    const float* __restrict__ W1,   // (256,128)
    const float* __restrict__ b1,   // (256,)
    const float* __restrict__ W2,   // (128,256)
    const float* __restrict__ b2,   // (128,)
    float* __restrict__ out)        // (3,2048,50,128)
{
    extern __shared__ float smem[];
    float* sW1 = smem + OFF_W1;
    float* sW2 = smem + OFF_W2;
    float* sb1 = smem + OFF_B1;
    float* sb2 = smem + OFF_B2;
    float* sy  = smem + OFF_Y;
    float* syi = smem + OFF_YIN;
    float* sk1 = smem + OFF_K1;
    float* sk2 = smem + OFF_K2;
    float* sk3 = smem + OFF_K3;
    float* sh  = smem + OFF_H;
    float* sts = smem + OFF_TS;

    const int tid  = threadIdx.x;
    const int lane = tid & 31;
    const int wave = tid >> 5;          // 0..7
    const int half = lane >> 4;         // 0 or 1
    const int l16  = lane & 15;
    const int row0 = blockIdx.x * 16;   // 16 batch rows per block

    // ---- stage weights into LDS (K-major), then biases / timesteps ----
    for (int i = tid; i < HID * DAUG; i += 256) {        // W1: [256][128]
        int n = i >> 7, k = i & 127;
        sW1[k * 256 + n] = W1[i];
    }
    for (int i = tid; i < DAUG * HID; i += 256) {        // W2: [128][256]
        int d = i >> 8, k = i & 255;
        sW2[k * 128 + d] = W2[i];
    }
    if (tid < 256) sb1[tid] = b1[tid];
    if (tid < 128) sb2[tid] = b2[tid];
    if (tid < TSTEPS) sts[tid] = ts[tid];

    // ---- load y0 (latents + zero augmentation), emit step 0 output ----
    for (int i = tid; i < 16 * DAUG; i += 256) {
        int m = i >> 7, d = i & 127;
        int gm = row0 + m;
        float v = (d < LAT) ? fp[gm * LAT + d] : 0.0f;
        sy[m * 129 + d]  = v;
        syi[m * 129 + d] = v;
        __builtin_nontemporal_store(v, &out[((long)gm * TSTEPS + 0) * DAUG + d]);
    }
    __syncthreads();

    const int nb0 = wave * 32;          // GEMM1: this wave's two N-tiles
    const int nb1 = nb0 + 16;
    const int db  = wave * 16;          // GEMM2: this wave's D-tile
    const int a1row = l16 * 129 + 2 * half;
    const int a2row = l16 * 257 + 2 * half;

    // ---- hoist this wave's weight fragments into persistent registers ----
    // GEMM1 B-fragments: B1[k][n]=W1[n][k]; frag v-th value has K = 4j + 2*half + v
    v2f w1f0[32], w1f1[32];
    #pragma unroll
    for (int j = 0; j < 32; ++j) {
        int krow = (4 * j + 2 * half) * 256;
        w1f0[j].x = sW1[krow +       nb0 + l16];
        w1f0[j].y = sW1[krow + 256 + nb0 + l16];
        w1f1[j].x = sW1[krow +       nb1 + l16];
        w1f1[j].y = sW1[krow + 256 + nb1 + l16];
    }
    // GEMM2 B-fragments: B2[k][d]=W2[d][k]
    v2f w2f[64];
    #pragma unroll
    for (int j = 0; j < 64; ++j) {
        int krow = (4 * j + 2 * half) * 128;
        w2f[j].x = sW2[krow +       db + l16];
        w2f[j].y = sW2[krow + 128 + db + l16];
    }
    const float bias1a = sb1[nb0 + l16];
    const float bias1b = sb1[nb1 + l16];
    const float bias2  = sb2[db + l16];

    for (int s = 0; s < TSTEPS - 1; ++s) {
        const float hstep = sts[s + 1] - sts[s];

        #pragma unroll 1
        for (int e = 0; e < 4; ++e) {
            // ================= GEMM1: h = tanh(yin @ W1^T + b1) =================
            v8f acc0 = {}; v8f acc1 = {};
            #pragma unroll
            for (int j = 0; j < 32; ++j) {
                v2f a;  a.x = syi[a1row + 4 * j];
                        a.y = syi[a1row + 4 * j + 1];
                acc0 = __builtin_amdgcn_wmma_f32_16x16x4_f32(
                           false, a, false, w1f0[j], (short)0, acc0, false, false);
                acc1 = __builtin_amdgcn_wmma_f32_16x16x4_f32(
                           false, a, false, w1f1[j], (short)0, acc1, false, false);
            }
            #pragma unroll
            for (int v = 0; v < 8; ++v) {
                int m = v + 8 * half;
                sh[m * 257 + nb0 + l16] = tanhf(acc0[v] + bias1a);
                sh[m * 257 + nb1 + l16] = tanhf(acc1[v] + bias1b);
            }
            __syncthreads();

            // ================= GEMM2: k = h @ W2^T + b2 =========================
            v8f accA = {}; v8f accB = {};
            #pragma unroll
            for (int j = 0; j < 64; j += 2) {
                v2f aA; aA.x = sh[a2row + 4 * j];
                        aA.y = sh[a2row + 4 * j + 1];
                accA = __builtin_amdgcn_wmma_f32_16x16x4_f32(
                           false, aA, false, w2f[j], (short)0, accA, false, false);
                v2f aB; aB.x = sh[a2row + 4 * j + 4];
                        aB.y = sh[a2row + 4 * j + 5];
                accB = __builtin_amdgcn_wmma_f32_16x16x4_f32(
                           false, aB, false, w2f[j + 1], (short)0, accB, false, false);
            }

            // ================= RK4(3/8) stage epilogue =========================
            #pragma unroll
            for (int v = 0; v < 8; ++v) {
                int m = v + 8 * half;
                int d = db + l16;
                int idx = m * 129 + d;
                float kv = accA[v] + accB[v] + bias2;
                if (e == 0) {
                    sk1[idx] = kv;
                    syi[idx] = sy[idx] + hstep * (1.0f / 3.0f) * kv;
                } else if (e == 1) {
                    sk2[idx] = kv;
                    syi[idx] = sy[idx] + hstep * (kv - (1.0f / 3.0f) * sk1[idx]);
                } else if (e == 2) {
                    sk3[idx] = kv;
                    syi[idx] = sy[idx] + hstep * (sk1[idx] - sk2[idx] + kv);
                } else {
                    float yn = sy[idx] + hstep * 0.125f *
                               (sk1[idx] + 3.0f * sk2[idx] + 3.0f * sk3[idx] + kv);
                    sy[idx]  = yn;
                    syi[idx] = yn;
                    __builtin_nontemporal_store(
                        yn, &out[((long)(row0 + m) * TSTEPS + (s + 1)) * DAUG + d]);
                }
            }
            __syncthreads();
        }
    }
}

extern "C" void kernel_launch(void* const* d_in, const int* in_sizes, int n_in,
                              void* d_out, int out_size, void* d_ws, size_t ws_size,
                              hipStream_t stream) {
    (void)in_sizes; (void)n_in; (void)d_ws; (void)ws_size; (void)out_size;
    const float* fp = (const float*)d_in[0];
    const float* ts = (const float*)d_in[1];
    const float* W1 = (const float*)d_in[2];
    const float* b1 = (const float*)d_in[3];
    const float* W2 = (const float*)d_in[4];
    const float* b2 = (const float*)d_in[5];
    float* out = (float*)d_out;

    const size_t ldsBytes = (size_t)LDS_FLOATS * sizeof(float);  // ~314 KB
    dim3 grid(MROWS / 16);   // 384 blocks, each owns 16 batch rows
    dim3 block(256);         // 8 wave32 waves
    DiffeqSolver_augmented_26809185861945_kernel<<<grid, block, ldsBytes, stream>>>(
        fp, ts, W1, b1, W2, b2, out);
}